// SocialLSTM_90108413870877
// MI455X (gfx1250) — compile-verified
//
#include <hip/hip_runtime.h>

typedef __attribute__((ext_vector_type(2))) float v2f;
typedef __attribute__((ext_vector_type(4))) float v4f;
typedef __attribute__((ext_vector_type(8))) float v8f;

#define T_HIST   50
#define HID      128
#define GATES    512
#define PRED_LEN 60
#define BTILE    16
#define HS       132   // padded LDS row stride for h/hid (floats)
#define WLS      132   // padded LDS row stride for W_hh rows (floats); bank step 4
#define QSTRIDE  (16 * WLS)          // floats between gate blocks of one wave
#define WSTRIDE  (4 * 16 * WLS)      // floats between waves' W slices

__device__ __forceinline__ float sigm(float x) { return 1.0f / (1.0f + __expf(-x)); }
__device__ __forceinline__ float tanh_fast(float x) {
  // robust tanh via exp: saturates correctly at +/-1 for large |x|
  return 1.0f - 2.0f / (__expf(2.0f * x) + 1.0f);
}

// --- Precompute rank-2 folded input projection ---------------------------
// M[n][d]   = sum_e W_ih[n][e] * W_emb[e][d]          (512 x 2)
// bgate[n]  = b_ih[n] + b_hh[n] + sum_e W_ih[n][e]*b_emb[e]
// bgate0[n] = b_ih[n] + b_hh[n]   (decoder step 0: prev embedding is exactly 0)
__global__ void precompute_rank2(const float* __restrict__ W_emb,
                                 const float* __restrict__ b_emb,
                                 const float* __restrict__ W_ih,
                                 const float* __restrict__ b_ih,
                                 const float* __restrict__ b_hh,
                                 float* __restrict__ Mx,
                                 float* __restrict__ bgate,
                                 float* __restrict__ bgate0) {
  int n = blockIdx.x * blockDim.x + threadIdx.x;
  if (n >= GATES) return;
  float m0 = 0.f, m1 = 0.f, be = 0.f;
  for (int e = 0; e < 64; ++e) {
    float w = W_ih[n * 64 + e];
    m0 += w * W_emb[e * 2 + 0];
    m1 += w * W_emb[e * 2 + 1];
    be += w * b_emb[e];
  }
  Mx[n * 2 + 0] = m0;
  Mx[n * 2 + 1] = m1;
  float bb = b_ih[n] + b_hh[n];
  bgate[n]  = bb + be;
  bgate0[n] = bb;
}

// --- Persistent LSTM kernel: one workgroup = 16 batch rows, all 110 steps ---
// W_hh lives in LDS, re-blocked by (wave, gate) so every B-tile ds_load_b64 in
// the GEMM uses ONE base VGPR + a 16-bit immediate offset (q*8448 + k*4 bytes).
// Wave w owns hidden cols [16w,16w+16); computes i/f/g/o 16x16 tiles with
// V_WMMA_F32_16X16X4_F32; c stays in 8 VGPRs for all 110 steps; h is
// double-buffered in LDS.
__global__ __launch_bounds__(256) void social_lstm_persistent(
    const float* __restrict__ history,  // (B, 50, 2)
    const float* __restrict__ W_hh,     // (512, 128)
    const float* __restrict__ W1,       // (128, 128)
    const float* __restrict__ b1,       // (128)
    const float* __restrict__ W2,       // (2, 128)
    const float* __restrict__ b2,       // (2)
    const float* __restrict__ Mx,       // (512, 2)
    const float* __restrict__ bgate,    // (512)
    const float* __restrict__ bgate0,   // (512)
    float* __restrict__ out) {          // (B, 60, 2)
  __shared__ float w_lds[GATES * WLS];     // W_hh re-blocked: ~270 KB
  __shared__ float h_lds[2 * BTILE * HS];  // double-buffered h
  __shared__ float hid_lds[BTILE * HS];    // relu(h@W1^T+b1) staging
  __shared__ float x_lds[BTILE][2];        // per-step rank-2 input (hist or pred)
  __shared__ float w2_lds[2 * HID];
  __shared__ float b2_lds[2];

  const int tid  = threadIdx.x;
  const int wave = tid >> 5;
  const int lane = tid & 31;
  const int lm   = lane & 15;   // C-tile: N; A-tile: M row; B-tile: N col
  const int lh   = lane >> 4;   // half-wave select (K pair / M+8)
  const int jcol = wave * 16 + lm;
  const int rowBase = blockIdx.x * BTILE;

  // ---- one-time LDS setup -------------------------------------------------
  // W_hh copy with (wave, gate) re-blocking:
  //   src row n = q*128 + w*16 + r  ->  dst block (w*4 + q)*16 + r
  for (int idx = tid; idx < GATES * (HID / 4); idx += 256) {
    int n  = idx >> 5;          // source row, 32 x 4-float chunks per row
    int k4 = (idx & 31) << 2;
    int q = n >> 7, w = (n >> 4) & 7, r = n & 15;
    int dstrow = (w * 4 + q) * 16 + r;
    *(v4f*)(w_lds + dstrow * WLS + k4) = *(const v4f*)(W_hh + n * HID + k4);
  }
  for (int i = tid; i < 2 * BTILE * HS; i += 256) h_lds[i] = 0.0f;
  for (int i = tid; i < 2 * HID; i += 256) w2_lds[i] = W2[i];
  if (tid < 2) b2_lds[tid] = b2[tid];

  // Per-lane constants for the rank-2 input contribution + gate bias
  float mx0[4], mx1[4], bgv[4], bg0v[4];
#pragma unroll
  for (int q = 0; q < 4; ++q) {
    int n = q * HID + jcol;
    mx0[q]  = Mx[n * 2 + 0];
    mx1[q]  = Mx[n * 2 + 1];
    bgv[q]  = bgate[n];
    bg0v[q] = bgate0[n];
  }
  const float b1v = b1[jcol];

  // Single B-tile base: all 4 gate blocks reachable via immediate offsets.
  const float* wbase = w_lds + wave * WSTRIDE + lm * WLS + 2 * lh;

  const v8f vzero = {0.f, 0.f, 0.f, 0.f, 0.f, 0.f, 0.f, 0.f};
  v8f cacc = vzero;  // cell state tile: VGPR-resident for all 110 steps
  int cur = 0;

  // ================= Encoder: 50 steps =================
  for (int t = 0; t < T_HIST; ++t) {
    __syncthreads();  // h_lds[cur] (and on t==0 the LDS setup) stable
    if (tid < 32) {   // stage this step's raw inputs (rank-2 path)
      int m = tid >> 1, dd = tid & 1;
      x_lds[m][dd] = history[((size_t)(rowBase + m) * T_HIST + t) * 2 + dd];
    }
    v8f acc[4] = {vzero, vzero, vzero, vzero};
    const float* hb = h_lds + cur * BTILE * HS + lm * HS + 2 * lh;
#pragma unroll 8
    for (int k = 0; k < HID; k += 4) {
      v2f a = *(const v2f*)(hb + k);  // A: h[lm][k+2*lh .. +1]
#pragma unroll
      for (int q = 0; q < 4; ++q) {
        v2f bm = *(const v2f*)(wbase + q * QSTRIDE + k);  // pure imm offset
        acc[q] = __builtin_amdgcn_wmma_f32_16x16x4_f32(
            false, a, false, bm, (short)0, acc[q], false, false);
      }
    }
    __syncthreads();  // all reads of h_lds[cur] done; x_lds visible
    const int nxt = cur ^ 1;
    float* hout = h_lds + nxt * BTILE * HS;
#pragma unroll
    for (int r = 0; r < 8; ++r) {
      int m = r + 8 * lh;  // C layout: lanes>=16 hold rows M+8
      float x0 = x_lds[m][0], x1 = x_lds[m][1];
      float iv = sigm(acc[0][r] + x0 * mx0[0] + x1 * mx1[0] + bgv[0]);
      float fv = sigm(acc[1][r] + x0 * mx0[1] + x1 * mx1[1] + bgv[1]);
      float gv = tanh_fast(acc[2][r] + x0 * mx0[2] + x1 * mx1[2] + bgv[2]);
      float ov = sigm(acc[3][r] + x0 * mx0[3] + x1 * mx1[3] + bgv[3]);
      float cc = fv * cacc[r] + iv * gv;
      cacc[r] = cc;
      hout[m * HS + jcol] = ov * tanh_fast(cc);
    }
    cur = nxt;
  }

  // Preload this wave's W1 slice into registers (16 rows x 128 k = 64 VGPRs).
  // Loaded here (not earlier) to keep encoder register pressure low.
  v2f w1reg[32];
  {
    const float* w1row = W1 + (size_t)jcol * HID + 2 * lh;
#pragma unroll
    for (int kk = 0; kk < 32; ++kk) w1reg[kk] = *(const v2f*)(w1row + 4 * kk);
  }

  // ================= Decoder: 60 autoregressive steps =================
  for (int dstep = 0; dstep < PRED_LEN; ++dstep) {
    __syncthreads();  // h_lds[cur] stable; x_lds holds prev pred (dstep>=1)
    v8f acc[4] = {vzero, vzero, vzero, vzero};
    const float* hb = h_lds + cur * BTILE * HS + lm * HS + 2 * lh;
#pragma unroll 8
    for (int k = 0; k < HID; k += 4) {
      v2f a = *(const v2f*)(hb + k);
#pragma unroll
      for (int q = 0; q < 4; ++q) {
        v2f bm = *(const v2f*)(wbase + q * QSTRIDE + k);
        acc[q] = __builtin_amdgcn_wmma_f32_16x16x4_f32(
            false, a, false, bm, (short)0, acc[q], false, false);
      }
    }
    __syncthreads();
    float bi[4];
#pragma unroll
    for (int q = 0; q < 4; ++q) bi[q] = (dstep == 0) ? bg0v[q] : bgv[q];
    const int nxt = cur ^ 1;
    float* hout = h_lds + nxt * BTILE * HS;
#pragma unroll
    for (int r = 0; r < 8; ++r) {
      int m = r + 8 * lh;
      float x0 = (dstep == 0) ? 0.f : x_lds[m][0];
      float x1 = (dstep == 0) ? 0.f : x_lds[m][1];
      float iv = sigm(acc[0][r] + x0 * mx0[0] + x1 * mx1[0] + bi[0]);
      float fv = sigm(acc[1][r] + x0 * mx0[1] + x1 * mx1[1] + bi[1]);
      float gv = tanh_fast(acc[2][r] + x0 * mx0[2] + x1 * mx1[2] + bi[2]);
      float ov = sigm(acc[3][r] + x0 * mx0[3] + x1 * mx1[3] + bi[3]);
      float cc = fv * cacc[r] + iv * gv;
      cacc[r] = cc;
      hout[m * HS + jcol] = ov * tanh_fast(cc);
    }
    cur = nxt;
    __syncthreads();  // new h visible for FC1 GEMM

    // hid = relu(h @ W1^T + b1): B entirely register-resident, A from LDS
    v8f hacc = vzero;
    const float* hb2 = h_lds + cur * BTILE * HS + lm * HS + 2 * lh;
#pragma unroll
    for (int kk = 0; kk < 32; ++kk) {
      v2f a = *(const v2f*)(hb2 + 4 * kk);
      hacc = __builtin_amdgcn_wmma_f32_16x16x4_f32(
          false, a, false, w1reg[kk], (short)0, hacc, false, false);
    }
#pragma unroll
    for (int r = 0; r < 8; ++r) {
      int m = r + 8 * lh;
      hid_lds[m * HS + jcol] = fmaxf(hacc[r] + b1v, 0.f);
    }
    __syncthreads();  // hid visible

    // pred = hid @ W2^T + b2  (D=2 -> 32 dot products, wave 0 only; no WMMA)
    if (tid < 32) {
      int m = tid & 15, dd = tid >> 4;
      float s = b2_lds[dd];
      const float* hr = hid_lds + m * HS;
      const float* wr = w2_lds + dd * HID;
#pragma unroll 8
      for (int j = 0; j < HID; ++j) s += hr[j] * wr[j];
      x_lds[m][dd] = s;  // feeds next step's rank-2 input path
      out[((size_t)(rowBase + m) * PRED_LEN + dstep) * 2 + dd] = s;
    }
    // next iteration's first barrier publishes x_lds
  }
}

extern "C" void kernel_launch(void* const* d_in, const int* in_sizes, int n_in,
                              void* d_out, int out_size, void* d_ws, size_t ws_size,
                              hipStream_t stream) {
  const float* history = (const float*)d_in[0];
  const float* W_emb   = (const float*)d_in[1];
  const float* b_emb   = (const float*)d_in[2];
  const float* W_ih    = (const float*)d_in[3];
  const float* W_hh    = (const float*)d_in[4];
  const float* b_ih    = (const float*)d_in[5];
  const float* b_hh    = (const float*)d_in[6];
  const float* W1      = (const float*)d_in[7];
  const float* b1      = (const float*)d_in[8];
  const float* W2      = (const float*)d_in[9];
  const float* b2      = (const float*)d_in[10];
  float* out = (float*)d_out;
  float* ws  = (float*)d_ws;
  float* Mx  = ws;         // 1024 floats
  float* bg  = ws + 1024;  // 512 floats
  float* bg0 = ws + 1536;  // 512 floats

  precompute_rank2<<<2, 256, 0, stream>>>(W_emb, b_emb, W_ih, b_ih, b_hh, Mx, bg, bg0);

  const int B = in_sizes[0] / (T_HIST * 2);
  social_lstm_persistent<<<B / BTILE, 256, 0, stream>>>(
      history, W_hh, W1, b1, W2, b2, Mx, bg, bg0, out);
  (void)n_in; (void)out_size; (void)ws_size;
}